// FrameGNNEncoder_10582799417745
// MI455X (gfx1250) — compile-verified
//
#include <hip/hip_runtime.h>
#include <hip/hip_bf16.h>

typedef __attribute__((ext_vector_type(2))) float v2f;
typedef __attribute__((ext_vector_type(8))) float v8f;

#define F 128          // feature / hidden dim (reference: IN = H = 128)
#define FP 130         // padded LDS row stride (dwords) to avoid bank conflicts

// ---------------------------------------------------------------- zero fill
__global__ void zero_kernel(float* __restrict__ p, long long n) {
    long long i = (long long)blockIdx.x * blockDim.x + threadIdx.x;
    long long stride = (long long)gridDim.x * blockDim.x;
    for (; i < n; i += stride) p[i] = 0.0f;
}

// ------------------------------------------------- edge scatter (mean prep)
// One wave32 per edge: lane l handles 4 contiguous floats (float4 gather,
// 4 native fp32 atomic adds). Optionally accumulates the in-degree count
// (only on the first layer; counts are reused for layer 2).
__global__ void aggregate_kernel(const float* __restrict__ feat,
                                 const long long* __restrict__ src,
                                 const long long* __restrict__ dst,
                                 float* __restrict__ agg,
                                 float* __restrict__ cnt,   // may be nullptr
                                 long long E) {
    long long wid = ((long long)blockIdx.x * blockDim.x + threadIdx.x) >> 5;
    int lane = threadIdx.x & 31;
    long long nwaves = ((long long)gridDim.x * blockDim.x) >> 5;
    for (long long e = wid; e < E; e += nwaves) {
        long long s = src[e];
        long long d = dst[e];
        const float4 v = *(const float4*)(feat + s * F + lane * 4);
        float* ap = agg + d * F + lane * 4;
        unsafeAtomicAdd(ap + 0, v.x);
        unsafeAtomicAdd(ap + 1, v.y);
        unsafeAtomicAdd(ap + 2, v.z);
        unsafeAtomicAdd(ap + 3, v.w);
        if (cnt != nullptr && lane == 0) unsafeAtomicAdd(cnt + d, 1.0f);
    }
}

// ------------------------------------- fused mean + dual-GEMM + norm + relu
// out[m, :] = relu( L2norm( (agg[m,:]/max(cnt,1)) @ Wl^T + bl + x[m,:] @ Wr^T ) )
// Block = 256 threads = 8 waves; block owns 16 full rows (nodes).
// Wave w computes the 16x16 tile at columns [16w, 16w+16) with
// V_WMMA_F32_16X16X4_F32, K-loop over 128 in steps of 4, two WMMAs per
// step (Wl with the mean tile, Wr with the root tile) into one accumulator.
__global__ __launch_bounds__(256)
void sage_fused_kernel(const float* __restrict__ agg,
                       const float* __restrict__ cnt,
                       const float* __restrict__ xin,
                       const float* __restrict__ Wl,   // [H, F] row-major
                       const float* __restrict__ bl,   // [H]
                       const float* __restrict__ Wr,   // [H, F] row-major
                       float* __restrict__ out,        // [N, H]
                       int N) {
    __shared__ float meanS[16 * FP];
    __shared__ float xS[16 * FP];
    __shared__ float rowss[16];
    __shared__ float rinv[16];

    const int tid = threadIdx.x;
    const int m0 = blockIdx.x * 16;

    // Stage mean (agg/clamped-count) and root-feature tiles into LDS.
    for (int i = tid; i < 16 * F; i += 256) {
        int r = i >> 7;         // local row 0..15
        int c = i & (F - 1);    // column 0..127
        int row = m0 + r; if (row >= N) row = N - 1;  // clamp (no divergence)
        float ic = 1.0f / fmaxf(cnt[row], 1.0f);
        meanS[r * FP + c] = agg[(long long)row * F + c] * ic;
        xS[r * FP + c]    = xin[(long long)row * F + c];
    }
    if (tid < 16) rowss[tid] = 0.0f;
    __syncthreads();

    const int lane = tid & 31;
    const int w    = tid >> 5;
    const int n0   = w * 16;           // output column base for this wave
    const int l16  = lane & 15;        // A-row / B-col / D-col index
    const int half = lane >> 4;        // selects K-pair and M-half

    const float* wlp = Wl + (long long)(n0 + l16) * F;  // row of Wl = column of B
    const float* wrp = Wr + (long long)(n0 + l16) * F;

    v8f acc = {};
    #pragma unroll
    for (int k = 0; k < F; k += 4) {
        const int ko = k + 2 * half;   // this lane's K offset (even -> 8B aligned)
        v2f am = *(const v2f*)(meanS + l16 * FP + ko);
        v2f bw = *(const v2f*)(wlp + ko);
        acc = __builtin_amdgcn_wmma_f32_16x16x4_f32(
                  false, am, false, bw, (short)0, acc, false, false);
        v2f ax = *(const v2f*)(xS + l16 * FP + ko);
        v2f br = *(const v2f*)(wrp + ko);
        acc = __builtin_amdgcn_wmma_f32_16x16x4_f32(
                  false, ax, false, br, (short)0, acc, false, false);
    }

    // Bias (indexed by output column), then per-row sum of squares.
    const float bias = bl[n0 + l16];
    #pragma unroll
    for (int v = 0; v < 8; v++) acc[v] += bias;

    // D layout: element v in lane -> row (v + 8*half), col l16.
    #pragma unroll
    for (int v = 0; v < 8; v++)
        unsafeAtomicAdd(&rowss[v + 8 * half], acc[v] * acc[v]);
    __syncthreads();

    if (tid < 16) {
        float nrm = sqrtf(rowss[tid]);
        rinv[tid] = 1.0f / fmaxf(nrm, 1e-12f);
    }
    __syncthreads();

    #pragma unroll
    for (int v = 0; v < 8; v++) {
        int r = v + 8 * half;
        int row = m0 + r;
        if (row < N) {
            float val = acc[v] * rinv[r];
            out[(long long)row * F + n0 + l16] = fmaxf(val, 0.0f);
        }
    }
}

// --------------------------------------------------------------------- host
extern "C" void kernel_launch(void* const* d_in, const int* in_sizes, int n_in,
                              void* d_out, int out_size, void* d_ws, size_t ws_size,
                              hipStream_t stream) {
    const float*     x   = (const float*)d_in[0];
    const long long* ei  = (const long long*)d_in[1];  // int64 [2, E]
    const float*     Wl0 = (const float*)d_in[2];
    const float*     bl0 = (const float*)d_in[3];
    const float*     Wr0 = (const float*)d_in[4];
    const float*     Wl1 = (const float*)d_in[5];
    const float*     bl1 = (const float*)d_in[6];
    const float*     Wr1 = (const float*)d_in[7];
    float* out = (float*)d_out;

    const int H  = in_sizes[3];            // 128
    const int IN = in_sizes[2] / H;        // 128 (== F)
    const int N  = in_sizes[0] / IN;       // 50000
    const long long E = in_sizes[1] / 2;   // 800000
    (void)H; (void)n_in; (void)out_size; (void)ws_size;

    // Workspace: [cnt: N floats (padded to 64)] [agg: N*F floats]
    float* cnt = (float*)d_ws;
    long long cntPad = ((long long)N + 63) & ~63LL;
    float* agg = cnt + cntPad;

    const long long src_off = 0, dst_off = E;
    const long long aggN = (long long)N * F;

    int zb_cnt = (int)((cntPad + 255) / 256);
    int zb_agg = (int)((aggN + 255) / 256);
    int agg_blocks = (int)((E * 32 + 255) / 256);   // one wave32 per edge
    int gemm_blocks = (N + 15) / 16;

    // ---- layer 0 ----
    zero_kernel<<<zb_cnt, 256, 0, stream>>>(cnt, cntPad);
    zero_kernel<<<zb_agg, 256, 0, stream>>>(agg, aggN);
    aggregate_kernel<<<agg_blocks, 256, 0, stream>>>(
        x, ei + src_off, ei + dst_off, agg, cnt, E);
    sage_fused_kernel<<<gemm_blocks, 256, 0, stream>>>(
        agg, cnt, x, Wl0, bl0, Wr0, out, N);        // h0 -> d_out

    // ---- layer 1 (in place on d_out; blocks only touch their own rows) ----
    zero_kernel<<<zb_agg, 256, 0, stream>>>(agg, aggN);
    aggregate_kernel<<<agg_blocks, 256, 0, stream>>>(
        out, ei + src_off, ei + dst_off, agg, nullptr, E);
    sage_fused_kernel<<<gemm_blocks, 256, 0, stream>>>(
        agg, cnt, out, Wl1, bl1, Wr1, out, N);
}